// F2DAgg_16604343566505
// MI455X (gfx1250) — compile-verified
//
#include <hip/hip_runtime.h>
#include <math.h>

// ---------------------------------------------------------------------------
// F2DAgg for MI455X (gfx1250, wave32, WMMA + async global->LDS copies).
// Memory-bound (~600 MB @ 23.3 TB/s >> 51 GFLOP f32 on the matrix pipe):
// keep full f32 precision (V_WMMA_F32_16X16X4_F32) and hide HBM latency with
// GLOBAL_LOAD_ASYNC_TO_LDS double buffering (ASYNCcnt / s_wait_asynccnt).
// Diagonal scalings are re-associated so every async tile is a raw byte copy:
//   T' = diag(d)*(FE @ (diag(d)*Abin))*diag(d)   (scales in B-tile + at store)
//   C  = diag-col(d) * (T' @ FE^T)               (scale at store)
// ---------------------------------------------------------------------------

typedef __attribute__((ext_vector_type(2))) float v2f;
typedef __attribute__((ext_vector_type(8))) float v8f;

#define Bq   64          // batch
#define Nn   512         // N
#define OUTC 256         // output channels
#define KB   16          // K-chunk per LDS stage
#define BM   64          // block tile rows
#define BN   64          // block tile cols
#define NTHR 128         // 4 waves of 32
#define LDA  20          // A-tile LDS stride (80B rows: conflict-free + 16B aligned)
#define LDB  80          // B-tile LDS stride (half-waves 32 banks apart, 16B aligned)

// ---- CDNA5 async global->LDS copies (ASYNCcnt-tracked) ---------------------
__device__ __forceinline__ void async_copy_b128(void* lds, const void* gptr) {
  unsigned l = (unsigned)(unsigned long long)lds;   // low 32b of generic = LDS offset
  asm volatile("global_load_async_to_lds_b128 %0, %1, off"
               :: "v"(l), "v"(gptr) : "memory");
}
__device__ __forceinline__ void async_copy_b32(void* lds, const void* gptr) {
  unsigned l = (unsigned)(unsigned long long)lds;
  asm volatile("global_load_async_to_lds_b32 %0, %1, off"
               :: "v"(l), "v"(gptr) : "memory");
}
#define ASYNC_WAIT(imm) asm volatile("s_wait_asynccnt " imm ::: "memory")

// ---- wave32 WMMA core: one K-chunk (KB=16) of a 32x32 per-wave tile --------
__device__ __forceinline__ void wmma_chunk(const float* __restrict__ As,
                                           const float* __restrict__ Bs,
                                           v8f acc[2][2],
                                           int lane, int waveM, int waveN) {
  const int half  = lane >> 4;        // 0: lanes 0-15, 1: lanes 16-31
  const int lidx  = lane & 15;
  const int mrow0 = waveM * 32 + lidx;
  const int ncol0 = waveN * 32 + lidx;
#pragma unroll
  for (int kk = 0; kk < KB; kk += 4) {
    const int kb = kk + (half << 1);  // K pair base: +0 (lo half) / +2 (hi half)
    v2f a[2], b[2];
#pragma unroll
    for (int mi = 0; mi < 2; ++mi) {
      const float* ap = As + (mrow0 + mi * 16) * LDA + kb;
      a[mi].x = ap[0];
      a[mi].y = ap[1];
    }
#pragma unroll
    for (int ni = 0; ni < 2; ++ni) {
      b[ni].x = Bs[(kb    ) * LDB + ncol0 + ni * 16];
      b[ni].y = Bs[(kb + 1) * LDB + ncol0 + ni * 16];
    }
#pragma unroll
    for (int mi = 0; mi < 2; ++mi)
#pragma unroll
      for (int ni = 0; ni < 2; ++ni)
        acc[mi][ni] = __builtin_amdgcn_wmma_f32_16x16x4_f32(
            false, a[mi], false, b[ni], (short)0, acc[mi][ni], false, false);
  }
}

// ---- kernel 1: d12[b,i] = rsqrt(sum_j FE[b,i,j]) ---------------------------
__global__ __launch_bounds__(256)
void rowsum_rsqrt_k(const float* __restrict__ FE, float* __restrict__ d12) {
  __shared__ float red[256];
  const int r = blockIdx.x;
  const float* row = FE + (size_t)r * Nn;
  float s = 0.f;
  for (int j = threadIdx.x; j < Nn; j += 256) s += row[j];
  red[threadIdx.x] = s;
  __syncthreads();
  for (int o = 128; o > 0; o >>= 1) {
    if (threadIdx.x < (unsigned)o) red[threadIdx.x] += red[threadIdx.x + o];
    __syncthreads();
  }
  if (threadIdx.x == 0) d12[r] = rsqrtf(red[0]);
}

// ---- kernel 2: T' = diag(d) * (FE @ (diag(d)*binarize(AM))) * diag(d) ------
// A tile: raw FE via async b128 (double buffered).  B tile: compare+scale in
// registers (software pipelined), ds_store.  Q_async per wave per chunk = 2.
__global__ __launch_bounds__(NTHR)
void gemm_SA(const float* __restrict__ FE, const float* __restrict__ AM,
             const float* __restrict__ d12, float* __restrict__ Tp) {
  __shared__ float As[2][BM * LDA];
  __shared__ float Bs[2][KB * LDB];
  const int b  = blockIdx.z;
  const int m0 = blockIdx.y * BM;
  const int n0 = blockIdx.x * BN;
  const float* FEb = FE + (size_t)b * Nn * Nn;
  const float* AMb = AM + (size_t)b * Nn * Nn;
  const float* dv  = d12 + b * Nn;
  const int tid = threadIdx.x, lane = tid & 31, wave = tid >> 5;
  const int waveM = wave >> 1, waveN = wave & 1;

  const int am_ = tid >> 2, ak_ = (tid & 3) * 4;  // A b128 slot (i adds m+=32)
  const int bk_ = tid >> 6, bn_ = tid & 63;       // B slot (i adds k+=2)

  auto issueA = [&](int k0, int sbuf) {
#pragma unroll
    for (int i = 0; i < 2; ++i) {
      const int m = am_ + i * 32;
      async_copy_b128(&As[sbuf][m * LDA + ak_],
                      &FEb[(size_t)(m0 + m) * Nn + (k0 + ak_)]);
    }
  };
  auto loadB = [&](int k0, float* br) {
#pragma unroll
    for (int i = 0; i < 8; ++i) {
      const int k = bk_ + i * 2;
      const float a = AMb[(size_t)(k0 + k) * Nn + (n0 + bn_)];
      br[i] = (a > 0.f) ? dv[k0 + k] : 0.f;       // fold d[k] into B
    }
  };

  float bcur[8], bnext[8];
  v8f acc[2][2] = {};
  issueA(0, 0);
  loadB(0, bcur);
  const int NC = Nn / KB;
  for (int c = 0; c < NC; ++c) {
    const int s = c & 1;
#pragma unroll
    for (int i = 0; i < 8; ++i)                   // stage B chunk c
      Bs[s][(bk_ + i * 2) * LDB + bn_] = bcur[i];
    if (c + 1 < NC) {
      issueA((c + 1) * KB, 1 - s);
      loadB((c + 1) * KB, bnext);
      ASYNC_WAIT("0x2");                          // chunk c done, c+1 in flight
    } else {
      ASYNC_WAIT("0x0");
    }
    __syncthreads();
    wmma_chunk(As[s], Bs[s], acc, lane, waveM, waveN);
    __syncthreads();
#pragma unroll
    for (int i = 0; i < 8; ++i) bcur[i] = bnext[i];
  }
  // store T' = d[m] * acc * d[n]
  float* Tb = Tp + (size_t)b * Nn * Nn;
  const int half = lane >> 4, colx = lane & 15;
#pragma unroll
  for (int mi = 0; mi < 2; ++mi)
#pragma unroll
    for (int ni = 0; ni < 2; ++ni) {
      const int gcol = n0 + waveN * 32 + ni * 16 + colx;
      const float sc = dv[gcol];
#pragma unroll
      for (int v = 0; v < 8; ++v) {
        const int grow = m0 + waveM * 32 + mi * 16 + half * 8 + v;
        Tb[(size_t)grow * Nn + gcol] = dv[grow] * acc[mi][ni][v] * sc;
      }
    }
}

// ---- kernel 3: C[i][j] = d[j] * (T' @ FE^T)[i][j] --------------------------
// A tile: raw T' async b128.  B tile: transposed FE gather via async b32.
// Q_async per wave per chunk = 2 + 8 = 10.
__global__ __launch_bounds__(NTHR)
void gemm_TSt(const float* __restrict__ Tp, const float* __restrict__ FE,
              const float* __restrict__ d12, float* __restrict__ C) {
  __shared__ float As[2][BM * LDA];
  __shared__ float Bs[2][KB * LDB];
  const int b  = blockIdx.z;
  const int m0 = blockIdx.y * BM;
  const int n0 = blockIdx.x * BN;
  const float* Tb  = Tp + (size_t)b * Nn * Nn;
  const float* FEb = FE + (size_t)b * Nn * Nn;
  const float* dv  = d12 + b * Nn;
  const int tid = threadIdx.x, lane = tid & 31, wave = tid >> 5;
  const int waveM = wave >> 1, waveN = wave & 1;

  const int am_ = tid >> 2, ak_ = (tid & 3) * 4;  // A b128 slot
  const int tk_ = tid & 15, tn_ = tid >> 4;       // B b32 slot (i adds n+=8)

  auto issueTiles = [&](int k0, int sbuf) {
#pragma unroll
    for (int i = 0; i < 2; ++i) {
      const int m = am_ + i * 32;
      async_copy_b128(&As[sbuf][m * LDA + ak_],
                      &Tb[(size_t)(m0 + m) * Nn + (k0 + ak_)]);
    }
#pragma unroll
    for (int i = 0; i < 8; ++i) {                 // Bs[k][n] = FE[n0+n][k0+k]
      const int n = tn_ + i * 8;
      async_copy_b32(&Bs[sbuf][tk_ * LDB + n],
                     &FEb[(size_t)(n0 + n) * Nn + (k0 + tk_)]);
    }
  };

  v8f acc[2][2] = {};
  issueTiles(0, 0);
  const int NC = Nn / KB;
  for (int c = 0; c < NC; ++c) {
    const int s = c & 1;
    if (c + 1 < NC) {
      issueTiles((c + 1) * KB, 1 - s);
      ASYNC_WAIT("0xa");                          // chunk c done, c+1 (10) in flight
    } else {
      ASYNC_WAIT("0x0");
    }
    __syncthreads();
    wmma_chunk(As[s], Bs[s], acc, lane, waveM, waveN);
    __syncthreads();
  }
  float* Cb = C + (size_t)b * Nn * Nn;
  const int half = lane >> 4, colx = lane & 15;
#pragma unroll
  for (int mi = 0; mi < 2; ++mi)
#pragma unroll
    for (int ni = 0; ni < 2; ++ni) {
      const int gcol = n0 + waveN * 32 + ni * 16 + colx;
      const float sc = dv[gcol];                  // column scale d[j]
#pragma unroll
      for (int v = 0; v < 8; ++v) {
        const int grow = m0 + waveM * 32 + mi * 16 + half * 8 + v;
        Cb[(size_t)grow * Nn + gcol] = acc[mi][ni][v] * sc;
      }
    }
}

// ---- kernel 4: alpha = sigmoid([FE|C]·Wa + ba); C <- alpha*C + (1-alpha)*FE
__global__ __launch_bounds__(256)
void alpha_gate_k(const float* __restrict__ FE, float* __restrict__ Cbuf,
                  const float* __restrict__ Wa, const float* __restrict__ ba) {
  __shared__ float red[256];
  __shared__ float s_alpha;
  const int r = blockIdx.x;
  const size_t off = (size_t)r * Nn;
  float part = 0.f;
  for (int j = threadIdx.x; j < Nn; j += 256)
    part += FE[off + j] * Wa[j] + Cbuf[off + j] * Wa[Nn + j];
  red[threadIdx.x] = part;
  __syncthreads();
  for (int o = 128; o > 0; o >>= 1) {
    if (threadIdx.x < (unsigned)o) red[threadIdx.x] += red[threadIdx.x + o];
    __syncthreads();
  }
  if (threadIdx.x == 0) s_alpha = 1.f / (1.f + __expf(-(red[0] + ba[0])));
  __syncthreads();
  const float al = s_alpha;
  for (int j = threadIdx.x; j < Nn; j += 256) {
    const float c = Cbuf[off + j], f = FE[off + j];
    Cbuf[off + j] = al * c + (1.f - al) * f;      // gated residual, in place
  }
}

// ---- kernel 5: out = LeakyReLU([fe|DN] @ Wp + bp) --------------------------
// K = 1024 split 512|512 chunk-aligned.  All tiles raw async b128, Q = 4.
__global__ __launch_bounds__(NTHR)
void gemm_out(const float* __restrict__ FEG, const float* __restrict__ DN,
              const float* __restrict__ Wp, const float* __restrict__ bp,
              float* __restrict__ out) {
  __shared__ float As[2][BM * LDA];
  __shared__ float Bs[2][KB * LDB];
  const int m0 = blockIdx.y * BM;                 // over B*N = 32768 rows
  const int n0 = blockIdx.x * BN;                 // over OUTC
  const int tid = threadIdx.x, lane = tid & 31, wave = tid >> 5;
  const int waveM = wave >> 1, waveN = wave & 1;

  const int am_ = tid >> 2, ak_ = (tid & 3) * 4;  // A b128 slot
  const int wk_ = tid >> 4, wn_ = (tid & 15) * 4; // B b128 slot (i adds k+=8)

  auto issueTiles = [&](int k0, int sbuf) {
    const float* Asrc = (k0 < Nn) ? FEG : DN;
    const int kadj = (k0 < Nn) ? k0 : (k0 - Nn);
#pragma unroll
    for (int i = 0; i < 2; ++i) {
      const int m = am_ + i * 32;
      async_copy_b128(&As[sbuf][m * LDA + ak_],
                      &Asrc[(size_t)(m0 + m) * Nn + (kadj + ak_)]);
    }
#pragma unroll
    for (int i = 0; i < 2; ++i) {
      const int k = wk_ + i * 8;
      async_copy_b128(&Bs[sbuf][k * LDB + wn_],
                      &Wp[(size_t)(k0 + k) * OUTC + (n0 + wn_)]);
    }
  };

  v8f acc[2][2] = {};
  issueTiles(0, 0);
  const int NC = (2 * Nn) / KB;
  for (int c = 0; c < NC; ++c) {
    const int s = c & 1;
    if (c + 1 < NC) {
      issueTiles((c + 1) * KB, 1 - s);
      ASYNC_WAIT("0x4");                          // chunk c done, c+1 (4) in flight
    } else {
      ASYNC_WAIT("0x0");
    }
    __syncthreads();
    wmma_chunk(As[s], Bs[s], acc, lane, waveM, waveN);
    __syncthreads();
  }
  // epilogue: + bias, LeakyReLU(0.01)
  const int half = lane >> 4, colx = lane & 15;
#pragma unroll
  for (int mi = 0; mi < 2; ++mi)
#pragma unroll
    for (int ni = 0; ni < 2; ++ni) {
      const int gcol = n0 + waveN * 32 + ni * 16 + colx;
      const float bias = bp[gcol];
#pragma unroll
      for (int v = 0; v < 8; ++v) {
        const int grow = m0 + waveM * 32 + mi * 16 + half * 8 + v;
        const float x = acc[mi][ni][v] + bias;
        out[(size_t)grow * OUTC + gcol] = (x >= 0.f) ? x : 0.01f * x;
      }
    }
}

// ---------------------------------------------------------------------------
extern "C" void kernel_launch(void* const* d_in, const int* in_sizes, int n_in,
                              void* d_out, int out_size, void* d_ws, size_t ws_size,
                              hipStream_t stream) {
  const float* FE = (const float*)d_in[0];   // feature_edge     [B,N,N]
  const float* DN = (const float*)d_in[1];   // distribution_node[B,N,N]
  const float* AM = (const float*)d_in[2];   // associated_matrix[B,N,N]
  const float* Wa = (const float*)d_in[3];   // [2N,1]
  const float* ba = (const float*)d_in[4];   // [1]
  const float* Wp = (const float*)d_in[5];   // [2N,OUTC]
  const float* bp = (const float*)d_in[6];   // [OUTC]
  float* out = (float*)d_out;                // [B,N,OUTC]

  // workspace: d12 (128 KB, 1 MB aligned slot) | T' (64 MB) | C (64 MB)
  char* w = (char*)d_ws;
  float* d12 = (float*)w;
  float* Tp  = (float*)(w + (size_t)(1u << 20));
  float* C   = (float*)(w + (size_t)(1u << 20) + (size_t)Bq * Nn * Nn * sizeof(float));

  rowsum_rsqrt_k<<<Bq * Nn, 256, 0, stream>>>(FE, d12);
  gemm_SA <<<dim3(Nn / BN, Nn / BM, Bq), NTHR, 0, stream>>>(FE, AM, d12, Tp);
  gemm_TSt<<<dim3(Nn / BN, Nn / BM, Bq), NTHR, 0, stream>>>(Tp, FE, d12, C);
  alpha_gate_k<<<Bq * Nn, 256, 0, stream>>>(FE, C, Wa, ba);
  gemm_out<<<dim3(OUTC / BN, (Bq * Nn) / BM), NTHR, 0, stream>>>(C, DN, Wp, bp, out);
}